// GNNLayer_70746701299878
// MI455X (gfx1250) — compile-verified
//
#include <hip/hip_runtime.h>
#include <hip/hip_bf16.h>

#define Nn 50000
#define Ee 800000
#define HX 64
#define HY 32
#define HT 32
#define FEAT 96           // HX + HY
#define KX 128            // HX + HY + HT
#define GROUPS ((Nn + 15) / 16)   // 3125 row-groups of 16 nodes

typedef __attribute__((ext_vector_type(2))) float v2f;
typedef __attribute__((ext_vector_type(8))) float v8f;

// ---------------------------------------------------------------------------
// Kernel 1: zero the aggregation workspace (N x 96 fp32)
// ---------------------------------------------------------------------------
__global__ void zero_ws(float* __restrict__ aggr, int n) {
    int i = blockIdx.x * blockDim.x + threadIdx.x;
    if (i < n) aggr[i] = 0.0f;
}

// ---------------------------------------------------------------------------
// Kernel 2: gather + scatter-add.  One WAVE per edge: lanes cover 32
// consecutive features per chunk (3 chunks = 96 features), so the gather
// from h_X/h_Y and the f32 atomic scatter are coalesced 128B bursts.
// unsafeAtomicAdd guarantees the hardware global_atomic_add_f32 path (no CAS
// loop); atomics resolve in L2 (aggr = 19.2MB << 192MB L2).
// Edge indices are read once per wave instead of once per feature.
// ---------------------------------------------------------------------------
__global__ void scatter_add(const int* __restrict__ erow,
                            const int* __restrict__ ecol,
                            const float* __restrict__ hX,
                            const float* __restrict__ hY,
                            float* __restrict__ aggr) {
    const int lane = threadIdx.x & 31;
    const int e    = blockIdx.x * 8 + (threadIdx.x >> 5);   // 8 waves/block
    if (e >= Ee) return;
    const int col = ecol[e];
    const int row = erow[e];
    const float* srcX = hX + (size_t)col * HX;
    const float* srcY = hY + (size_t)col * HY;
    float* dst = aggr + (size_t)row * FEAT;
    unsafeAtomicAdd(&dst[lane],      srcX[lane]);        // features  0..31
    unsafeAtomicAdd(&dst[lane + 32], srcX[lane + 32]);   // features 32..63
    unsafeAtomicAdd(&dst[lane + 64], srcY[lane]);        // features 64..95
}

// ---------------------------------------------------------------------------
// fp32 WMMA helpers.  V_WMMA_F32_16X16X4_F32 lane layout:
//   A (16x4): lane l holds A[l%16][2*(l/16) + v], v = 0..1
//   B (4x16): lane l holds B[2*(l/16) + v][l%16]
//   C/D (16x16): lane l, VGPR v -> row v + 8*(l/16), col l%16
// ---------------------------------------------------------------------------
__device__ __forceinline__ v8f wmma_f32(v2f a, v2f b, v8f c) {
    return __builtin_amdgcn_wmma_f32_16x16x4_f32(
        /*neg_a=*/false, a, /*neg_b=*/false, b,
        /*c_mod=*/(short)0, c, /*reuse_a=*/false, /*reuse_b=*/false);
}

// ---------------------------------------------------------------------------
// Kernel 3: new_X = LayerNorm(relu([aggr | h_t] @ W_X + b_X))
// One wave computes a 16(node) x 64(col) tile: 4 WMMA accumulators, K loop
// over 128 in steps of 4.  Result bounced through LDS for the row-wise
// LayerNorm (2 lanes per row, __shfl_xor to combine halves).
// 4 waves per block, each owning a disjoint 16-node group + LDS slab.
// ---------------------------------------------------------------------------
__global__ void gemm_ln_X(const float* __restrict__ aggr,
                          const float* __restrict__ ht,
                          const float* __restrict__ WX,
                          const float* __restrict__ bX,
                          const float* __restrict__ gX,
                          const float* __restrict__ beX,
                          float* __restrict__ outX) {
    __shared__ float lds[4][16][HX + 1];   // +1 pad vs bank conflicts
    const int wave  = threadIdx.x >> 5;
    const int lane  = threadIdx.x & 31;
    const int group = blockIdx.x * 4 + wave;

    if (group < GROUPS) {
        const int m     = lane & 15;
        const int khalf = lane >> 4;          // 0 or 1
        const int node  = group * 16 + m;
        const float* arow = aggr + (size_t)node * FEAT;

        v8f acc[4] = {};
        #pragma unroll 4
        for (int k0 = 0; k0 < KX; k0 += 4) {
            const int ka = k0 + khalf * 2;
            v2f a;
            if (ka < FEAT) { a.x = arow[ka];      a.y = arow[ka + 1]; }
            else           { a.x = ht[ka - FEAT]; a.y = ht[ka - FEAT + 1]; }
            #pragma unroll
            for (int t = 0; t < 4; ++t) {
                const int n = t * 16 + m;
                v2f b;
                b.x = WX[ka * HX + n];
                b.y = WX[(ka + 1) * HX + n];
                acc[t] = wmma_f32(a, b, acc[t]);
            }
        }
        // bias + relu, spill tile to LDS in (row, col) order
        #pragma unroll
        for (int t = 0; t < 4; ++t) {
            const int n = t * 16 + m;
            const float bias = bX[n];
            #pragma unroll
            for (int v = 0; v < 8; ++v) {
                const int row = khalf * 8 + v;
                float x = acc[t][v] + bias;
                lds[wave][row][n] = fmaxf(x, 0.0f);
            }
        }
    }
    __syncthreads();
    if (group < GROUPS) {
        const int row  = lane >> 1;       // 16 rows, 2 lanes each
        const int half = lane & 1;        // each lane sums 32 of 64 cols
        float s = 0.0f, s2 = 0.0f;
        #pragma unroll
        for (int j = 0; j < 32; ++j) {
            float x = lds[wave][row][half * 32 + j];
            s += x; s2 += x * x;
        }
        s  += __shfl_xor(s, 1);
        s2 += __shfl_xor(s2, 1);
        const float mean = s * (1.0f / 64.0f);
        const float var  = s2 * (1.0f / 64.0f) - mean * mean;
        const float inv  = rsqrtf(var + 1e-5f);
        const int node   = group * 16 + row;
        #pragma unroll
        for (int j = 0; j < 32; ++j) {
            const int c = half * 32 + j;
            float x = lds[wave][row][c];
            outX[(size_t)node * HX + c] = (x - mean) * inv * gX[c] + beX[c];
        }
    }
}

// ---------------------------------------------------------------------------
// Kernel 4: new_Y = LayerNorm(relu(aggr[:,64:96] @ W_Y + b_Y))
// Same structure, 16x32 tile per wave, K = 32.
// ---------------------------------------------------------------------------
__global__ void gemm_ln_Y(const float* __restrict__ aggr,
                          const float* __restrict__ WY,
                          const float* __restrict__ bY,
                          const float* __restrict__ gY,
                          const float* __restrict__ beY,
                          float* __restrict__ outY) {
    __shared__ float lds[4][16][HY + 1];
    const int wave  = threadIdx.x >> 5;
    const int lane  = threadIdx.x & 31;
    const int group = blockIdx.x * 4 + wave;

    if (group < GROUPS) {
        const int m     = lane & 15;
        const int khalf = lane >> 4;
        const int node  = group * 16 + m;
        const float* arow = aggr + (size_t)node * FEAT + HX;   // Y slice

        v8f acc[2] = {};
        #pragma unroll
        for (int k0 = 0; k0 < HY; k0 += 4) {
            const int ka = k0 + khalf * 2;
            v2f a;
            a.x = arow[ka];
            a.y = arow[ka + 1];
            #pragma unroll
            for (int t = 0; t < 2; ++t) {
                const int n = t * 16 + m;
                v2f b;
                b.x = WY[ka * HY + n];
                b.y = WY[(ka + 1) * HY + n];
                acc[t] = wmma_f32(a, b, acc[t]);
            }
        }
        #pragma unroll
        for (int t = 0; t < 2; ++t) {
            const int n = t * 16 + m;
            const float bias = bY[n];
            #pragma unroll
            for (int v = 0; v < 8; ++v) {
                const int row = khalf * 8 + v;
                float x = acc[t][v] + bias;
                lds[wave][row][n] = fmaxf(x, 0.0f);
            }
        }
    }
    __syncthreads();
    if (group < GROUPS) {
        const int row  = lane >> 1;
        const int half = lane & 1;        // each lane sums 16 of 32 cols
        float s = 0.0f, s2 = 0.0f;
        #pragma unroll
        for (int j = 0; j < 16; ++j) {
            float x = lds[wave][row][half * 16 + j];
            s += x; s2 += x * x;
        }
        s  += __shfl_xor(s, 1);
        s2 += __shfl_xor(s2, 1);
        const float mean = s * (1.0f / 32.0f);
        const float var  = s2 * (1.0f / 32.0f) - mean * mean;
        const float inv  = rsqrtf(var + 1e-5f);
        const int node   = group * 16 + row;
        #pragma unroll
        for (int j = 0; j < 16; ++j) {
            const int c = half * 16 + j;
            float x = lds[wave][row][c];
            outY[(size_t)node * HY + c] = (x - mean) * inv * gY[c] + beY[c];
        }
    }
}

// ---------------------------------------------------------------------------
extern "C" void kernel_launch(void* const* d_in, const int* in_sizes, int n_in,
                              void* d_out, int out_size, void* d_ws, size_t ws_size,
                              hipStream_t stream) {
    const int*   erow = (const int*)d_in[0];
    const int*   ecol = (const int*)d_in[1];
    const float* hX   = (const float*)d_in[2];
    const float* hY   = (const float*)d_in[3];
    const float* ht   = (const float*)d_in[4];
    const float* WX   = (const float*)d_in[5];
    const float* bX   = (const float*)d_in[6];
    const float* gX   = (const float*)d_in[7];
    const float* beX  = (const float*)d_in[8];
    const float* WY   = (const float*)d_in[9];
    const float* bY   = (const float*)d_in[10];
    const float* gY   = (const float*)d_in[11];
    const float* beY  = (const float*)d_in[12];

    float* out  = (float*)d_out;
    float* aggr = (float*)d_ws;                 // N * 96 fp32 = 19.2 MB

    const int nAggr = Nn * FEAT;
    zero_ws<<<(nAggr + 255) / 256, 256, 0, stream>>>(aggr, nAggr);

    // one wave per edge, 8 waves (256 threads) per block
    scatter_add<<<(Ee + 7) / 8, 256, 0, stream>>>(erow, ecol, hX, hY, aggr);

    const int blocks = (GROUPS + 3) / 4;        // 4 waves/block, 1 group/wave
    gemm_ln_X<<<blocks, 128, 0, stream>>>(aggr, ht, WX, bX, gX, beX, out);
    gemm_ln_Y<<<blocks, 128, 0, stream>>>(aggr, WY, bY, gY, beY, out + (size_t)Nn * HX);
}